// GNN_76244259439066
// MI455X (gfx1250) — compile-verified
//
#include <hip/hip_runtime.h>

typedef __attribute__((ext_vector_type(2))) float v2f;
typedef __attribute__((ext_vector_type(8))) float v8f;

// ---------------------------------------------------------------------------
// Degree / normalization precompute
// ---------------------------------------------------------------------------
__global__ void k_fill1(float* __restrict__ p, int n) {
    int i = blockIdx.x * blockDim.x + threadIdx.x;
    if (i < n) p[i] = 1.0f;            // self-loop contributes 1 to every degree
}

__global__ void k_deg(const int* __restrict__ dst, float* __restrict__ deg, int E) {
    int i = blockIdx.x * blockDim.x + threadIdx.x;
    if (i < E) unsafeAtomicAdd(&deg[dst[i]], 1.0f);   // global_atomic_add_f32
}

__global__ void k_rsqrt_inplace(float* __restrict__ d, int n) {
    int i = blockIdx.x * blockDim.x + threadIdx.x;
    if (i < n) d[i] = rsqrtf(d[i]);    // deg >= 1 always (self-loops)
}

__global__ void k_norm(const int* __restrict__ src, const int* __restrict__ dst,
                       const float* __restrict__ dn, float* __restrict__ nrm, int E) {
    int i = blockIdx.x * blockDim.x + threadIdx.x;
    if (i < E) nrm[i] = dn[src[i]] * dn[dst[i]];
}

// ---------------------------------------------------------------------------
// Zero-pad x[N,11] -> XP[N,12]  (K multiple of 4 => branch-free GEMM loads)
// ---------------------------------------------------------------------------
__global__ void k_padX(const float* __restrict__ x, float* __restrict__ xp, int total) {
    int i = blockIdx.x * blockDim.x + threadIdx.x;
    if (i < total) {
        int n = i / 12, f = i - n * 12;
        xp[i] = (f < 11) ? x[n * 11 + f] : 0.0f;
    }
}

// ---------------------------------------------------------------------------
// Pre-pack W[K,NC] into WMMA B-fragment order, zero-padded to KP x NCP:
//   Wp[(pair*NCP + col)*2 + sub] = W[pair*2+sub][col]   (pair = k/2)
// Each lane's per-step B fragment is then one contiguous 8-byte pair.
// ---------------------------------------------------------------------------
template <int K, int NC, int KP, int NCP>
__global__ void k_packW(const float* __restrict__ W, float* __restrict__ Wp) {
    int i = blockIdx.x * blockDim.x + threadIdx.x;
    if (i < KP * NCP) {
        int pair = i / (NCP * 2);
        int rem  = i - pair * (NCP * 2);
        int col  = rem >> 1;
        int k    = pair * 2 + (rem & 1);
        Wp[i] = (k < K && col < NC) ? W[k * NC + col] : 0.0f;
    }
}

// ---------------------------------------------------------------------------
// Dense transform: C[M,NC] = A[M,KP] @ Wp, exact f32 via V_WMMA_F32_16X16X4_F32.
// One wave per 16x16 output tile; all hot-loop loads unconditional b64.
// A padded so KP % 4 == 0; Wp padded so NCP % 16 == 0. M % 16 == 0 (100000 ok).
// ---------------------------------------------------------------------------
template <int KP, int NCP, int NC>
__global__ __launch_bounds__(256) void k_gemm_wmma(const float* __restrict__ A,
                                                   const float* __restrict__ Wp,
                                                   float* __restrict__ C, int M) {
    constexpr int NT = NCP / 16;
    int wave = (blockIdx.x * blockDim.x + threadIdx.x) >> 5;   // uniform per wave
    int lane = threadIdx.x & 31;
    int totalWaves = (M >> 4) * NT;
    if (wave >= totalWaves) return;                            // wave-uniform branch

    int tM   = wave / NT;
    int tN   = wave - tM * NT;
    int half = lane >> 4;      // 0: lanes 0-15 (K+0,K+1), 1: lanes 16-31 (K+2,K+3)
    int lid  = lane & 15;
    int row  = (tM << 4) + lid;        // A row for this lane (M = lane&15)
    int col  = (tN << 4) + lid;        // B col for this lane (N = lane&15)

    const float* aPtr = A + (size_t)row * KP + half * 2;       // kb = kk + 2*half
    const float* bPtr = Wp + ((size_t)half * NCP + col) * 2;   // pair = kk/2 + half

    v8f acc = {};
#pragma unroll
    for (int kk = 0; kk < KP; kk += 4) {
        v2f a = *(const v2f*)(aPtr + kk);                  // global_load_b64
        v2f b = *(const v2f*)(bPtr + (kk >> 1) * NCP * 2); // global_load_b64
        acc = __builtin_amdgcn_wmma_f32_16x16x4_f32(
            false, a, false, b, (short)0, acc, false, false);
    }

    int rbase = (tM << 4) + half * 8;   // D VGPR j -> M = j + 8*half, N = lane&15
    if constexpr (NC == NCP) {
#pragma unroll
        for (int j = 0; j < 8; ++j) C[(rbase + j) * NC + col] = acc[j];
    } else {
        if (col < NC) {
#pragma unroll
            for (int j = 0; j < 8; ++j) C[(rbase + j) * NC + col] = acc[j];
        }
    }
}

// ---------------------------------------------------------------------------
// agg[n, f..f+3] = tmp[n, f..f+3] * dn[n]^2   (self-loop message, float4)
// ---------------------------------------------------------------------------
template <int SHIFT>
__global__ void k_selfinit4(const float* __restrict__ tmp, const float* __restrict__ dn,
                            float* __restrict__ agg, int total4) {
    int i = blockIdx.x * blockDim.x + threadIdx.x;
    if (i < total4) {
        float s = dn[i >> (SHIFT - 2)];
        s *= s;
        float4 v = *(const float4*)(tmp + i * 4);
        float4 r = make_float4(v.x * s, v.y * s, v.z * s, v.w * s);
        *(float4*)(agg + i * 4) = r;
    }
}

// ---------------------------------------------------------------------------
// agg[dst, f..f+3] += tmp[src, f..f+3] * norm[e]  — 4 features per thread:
// b128 feature load + 4x global_atomic_add_f32, edge indices amortized 4x.
// ---------------------------------------------------------------------------
template <int SHIFT>
__global__ void k_scatter4(const int* __restrict__ src, const int* __restrict__ dst,
                           const float* __restrict__ nrm, const float* __restrict__ tmp,
                           float* __restrict__ agg, int E) {
    constexpr int GS = SHIFT - 2;            // feature-groups per edge = 1<<GS
    int total  = E << GS;
    int stride = gridDim.x * blockDim.x;
    for (int i = blockIdx.x * blockDim.x + threadIdx.x; i < total; i += stride) {
        int e = i >> GS;
        int f = (i & ((1 << GS) - 1)) << 2;
        int s = src[e], d = dst[e];
        float w = nrm[e];
        float4 v = *(const float4*)(tmp + (s << SHIFT) + f);   // global_load_b128
        float* base = agg + (d << SHIFT) + f;
        unsafeAtomicAdd(base + 0, v.x * w);
        unsafeAtomicAdd(base + 1, v.y * w);
        unsafeAtomicAdd(base + 2, v.z * w);
        unsafeAtomicAdd(base + 3, v.w * w);
    }
}

// ---------------------------------------------------------------------------
// out[n, f..f+3] = act(agg[n, f..f+3] + bias[f..f+3])   (float4)
// ---------------------------------------------------------------------------
template <int SHIFT, bool RELU>
__global__ void k_bias_act4(const float* __restrict__ agg, const float* __restrict__ bias,
                            float* __restrict__ out, int total4) {
    int i = blockIdx.x * blockDim.x + threadIdx.x;
    if (i < total4) {
        int f = (i & ((1 << (SHIFT - 2)) - 1)) << 2;
        float4 b = *(const float4*)(bias + f);
        float4 v = *(const float4*)(agg + i * 4);
        float4 r = make_float4(v.x + b.x, v.y + b.y, v.z + b.z, v.w + b.w);
        if (RELU) {
            r.x = fmaxf(r.x, 0.f); r.y = fmaxf(r.y, 0.f);
            r.z = fmaxf(r.z, 0.f); r.w = fmaxf(r.w, 0.f);
        }
        *(float4*)(out + i * 4) = r;
    }
}

// ---------------------------------------------------------------------------
extern "C" void kernel_launch(void* const* d_in, const int* in_sizes, int n_in,
                              void* d_out, int out_size, void* d_ws, size_t ws_size,
                              hipStream_t stream) {
    const float* x  = (const float*)d_in[0];
    const int*   ei = (const int*)d_in[1];
    const float* W1 = (const float*)d_in[2]; const float* b1 = (const float*)d_in[3];
    const float* W2 = (const float*)d_in[4]; const float* b2 = (const float*)d_in[5];
    const float* W3 = (const float*)d_in[6]; const float* b3 = (const float*)d_in[7];
    const float* W4 = (const float*)d_in[8]; const float* b4 = (const float*)d_in[9];
    float* out = (float*)d_out;

    const int N = in_sizes[0] / 11;
    const int E = in_sizes[1] / 2;
    const int* srcI = ei;        // edge_index[0]
    const int* dstI = ei + E;    // edge_index[1]

    // ws (floats): B0[N*64] | B1[N*64] | norm[E] | dn[N] | XP[N*12] | Wp[4096]  ~69 MB
    float* B0  = (float*)d_ws;
    float* B1  = B0  + (size_t)N * 64;
    float* nrm = B1  + (size_t)N * 64;
    float* dn  = nrm + (size_t)E;
    float* XP  = dn  + (size_t)N;
    float* Wp  = XP  + (size_t)N * 12;

    const int TPB = 256;
    auto cdiv = [](int a, int b) { return (a + b - 1) / b; };
    const int SCAT_BLOCKS = 8192;

    // ---- normalization precompute (once, reused by all 4 layers) ----
    k_fill1<<<cdiv(N, TPB), TPB, 0, stream>>>(dn, N);
    k_deg<<<cdiv(E, TPB), TPB, 0, stream>>>(dstI, dn, E);
    k_rsqrt_inplace<<<cdiv(N, TPB), TPB, 0, stream>>>(dn, N);
    k_norm<<<cdiv(E, TPB), TPB, 0, stream>>>(srcI, dstI, dn, nrm, E);
    k_padX<<<cdiv(N * 12, TPB), TPB, 0, stream>>>(x, XP, N * 12);

    // ---- Layer 1: XP[N,12] @ W1(11,64) ; tmp=B0, agg=B1, h1=B0 ----
    {
        int waves = (N >> 4) * 4, nt4 = (N << 6) >> 2;
        k_packW<11, 64, 12, 64><<<cdiv(12 * 64, TPB), TPB, 0, stream>>>(W1, Wp);
        k_gemm_wmma<12, 64, 64><<<cdiv(waves * 32, TPB), TPB, 0, stream>>>(XP, Wp, B0, N);
        k_selfinit4<6><<<cdiv(nt4, TPB), TPB, 0, stream>>>(B0, dn, B1, nt4);
        k_scatter4<6><<<SCAT_BLOCKS, TPB, 0, stream>>>(srcI, dstI, nrm, B0, B1, E);
        k_bias_act4<6, true><<<cdiv(nt4, TPB), TPB, 0, stream>>>(B1, b1, B0, nt4);
    }
    // ---- Layer 2: h1[N,64] @ W2(64,32) ; tmp=B1, agg=B0, h2=B1 ----
    {
        int waves = (N >> 4) * 2, nt4 = (N << 5) >> 2;
        k_packW<64, 32, 64, 32><<<cdiv(64 * 32, TPB), TPB, 0, stream>>>(W2, Wp);
        k_gemm_wmma<64, 32, 32><<<cdiv(waves * 32, TPB), TPB, 0, stream>>>(B0, Wp, B1, N);
        k_selfinit4<5><<<cdiv(nt4, TPB), TPB, 0, stream>>>(B1, dn, B0, nt4);
        k_scatter4<5><<<SCAT_BLOCKS, TPB, 0, stream>>>(srcI, dstI, nrm, B1, B0, E);
        k_bias_act4<5, true><<<cdiv(nt4, TPB), TPB, 0, stream>>>(B0, b2, B1, nt4);
    }
    // ---- Layer 3: h2[N,32] @ W3(32,16) ; tmp=B0, agg=B1, h3=B0 ----
    {
        int waves = (N >> 4) * 1, nt4 = (N << 4) >> 2;
        k_packW<32, 16, 32, 16><<<cdiv(32 * 16, TPB), TPB, 0, stream>>>(W3, Wp);
        k_gemm_wmma<32, 16, 16><<<cdiv(waves * 32, TPB), TPB, 0, stream>>>(B1, Wp, B0, N);
        k_selfinit4<4><<<cdiv(nt4, TPB), TPB, 0, stream>>>(B0, dn, B1, nt4);
        k_scatter4<4><<<SCAT_BLOCKS, TPB, 0, stream>>>(srcI, dstI, nrm, B0, B1, E);
        k_bias_act4<4, true><<<cdiv(nt4, TPB), TPB, 0, stream>>>(B1, b3, B0, nt4);
    }
    // ---- Layer 4: h3[N,16] @ W4(16,8) ; tmp=B1, agg=B0, out (no ReLU) ----
    {
        int waves = (N >> 4) * 1, nt4 = (N << 3) >> 2;
        k_packW<16, 8, 16, 16><<<cdiv(16 * 16, TPB), TPB, 0, stream>>>(W4, Wp);
        k_gemm_wmma<16, 16, 8><<<cdiv(waves * 32, TPB), TPB, 0, stream>>>(B0, Wp, B1, N);
        k_selfinit4<3><<<cdiv(nt4, TPB), TPB, 0, stream>>>(B1, dn, B0, nt4);
        k_scatter4<3><<<SCAT_BLOCKS, TPB, 0, stream>>>(srcI, dstI, nrm, B1, B0, E);
        k_bias_act4<3, false><<<cdiv(nt4, TPB), TPB, 0, stream>>>(B0, b4, out, nt4);
    }
}